// ContrastMaximization_81853486727307
// MI455X (gfx1250) — compile-verified
//
#include <hip/hip_runtime.h>

// Problem constants (fixed by reference setup_inputs)
#define BATCH 2
#define NEV   262144
#define BASEB 10
#define RREF  11        // BASE + 1 reference times
#define HH    256
#define WW    256
#define EPSF  1e-9f
#define IMG   (HH * WW)                       // 65536
#define SZ    (BATCH * 2 * RREF * IMG)        // 2,883,584 floats per image array
#define TILES 16                              // reduction tiles per (b,r)

typedef float v8f __attribute__((ext_vector_type(8)));
typedef float v2f __attribute__((ext_vector_type(2)));

static __device__ __forceinline__ int iclamp(int v, int lo, int hi) {
  return v < lo ? lo : (v > hi ? hi : v);
}

static __device__ __forceinline__ void fadd_agent(float* p, float v) {
  // Relaxed, device(agent)-scope fadd -> native global_atomic_add_f32 (no CAS loop)
  __hip_atomic_fetch_add(p, v, __ATOMIC_RELAXED, __HIP_MEMORY_SCOPE_AGENT);
}

// ---------------------------------------------------------------------------
// Kernel 0: zero workspace (iwe, iwt, accumulators)
// ---------------------------------------------------------------------------
__global__ void cm_zero_kernel(float* __restrict__ ws, int n) {
  int stride = gridDim.x * blockDim.x;
  for (int i = blockIdx.x * blockDim.x + threadIdx.x; i < n; i += stride)
    ws[i] = 0.0f;
}

// ---------------------------------------------------------------------------
// Kernel 1: per-event flow gather + warp + bilinear scatter-add (L2 atomics)
// ---------------------------------------------------------------------------
__global__ __launch_bounds__(256) void cm_scatter_kernel(
    const float* __restrict__ events,     // [B, N, 5]
    const float* __restrict__ flow,       // [B, BASE, H, W, 2]
    float* __restrict__ iwe,              // [B*2*R*H*W]
    float* __restrict__ iwt) {
  int gid = blockIdx.x * blockDim.x + threadIdx.x;
  if (gid >= BATCH * NEV) return;
  int b = gid / NEV;

  const float* e = events + gid * 5;
  float x  = e[0];
  float y  = e[1];
  float t  = e[2];
  float ts = e[3];
  float p  = e[4];

  // --- bilinear flow sample at (x, y) in bin zi -----------------------------
  int zi = iclamp((int)floorf(t), 0, BASEB - 1);
  int x0 = iclamp((int)floorf(x), 0, WW - 2);
  int y0 = iclamp((int)floorf(y), 0, HH - 2);
  float fx = fminf(fmaxf(x - (float)x0, 0.0f), 1.0f);
  float fy = fminf(fmaxf(y - (float)y0, 0.0f), 1.0f);

  const float* fb = flow + (((b * BASEB + zi) * HH + y0) * WW + x0) * 2;
  float f00x = fb[0],           f00y = fb[1];
  float f01x = fb[2],           f01y = fb[3];
  float f10x = fb[2 * WW],      f10y = fb[2 * WW + 1];
  float f11x = fb[2 * WW + 2],  f11y = fb[2 * WW + 3];

  float w00 = (1.0f - fx) * (1.0f - fy);
  float w01 = fx * (1.0f - fy);
  float w10 = (1.0f - fx) * fy;
  float w11 = fx * fy;
  float flx = w00 * f00x + w01 * f01x + w10 * f10x + w11 * f11x;
  float fly = w00 * f00y + w01 * f01y + w10 * f10y + w11 * f11y;

  int pidx = (p > 0.5f) ? 1 : 0;
  int chanbase = (b * 2 + pidx) * RREF;

  // --- warp to each reference time, bilinear scatter ------------------------
  for (int r = 0; r < RREF; ++r) {
    float dt = (float)r - t;
    float wx = x + dt * flx;
    float wy = y + dt * fly;
    float fwx = floorf(wx);
    float fwy = floorf(wy);
    int X0 = (int)fwx;
    int Y0 = (int)fwy;
    float gx = wx - fwx;
    float gy = wy - fwy;
    int base = (chanbase + r) * IMG;

    float cw[4] = { (1.0f - gx) * (1.0f - gy), gx * (1.0f - gy),
                    (1.0f - gx) * gy,          gx * gy };
    int cx[4] = { X0, X0 + 1, X0,     X0 + 1 };
    int cy[4] = { Y0, Y0,     Y0 + 1, Y0 + 1 };
    #pragma unroll
    for (int c = 0; c < 4; ++c) {
      int xi = cx[c], yi = cy[c];
      if (xi >= 0 && xi < WW && yi >= 0 && yi < HH) {
        int idx = base + yi * WW + xi;
        float wv = cw[c];
        fadd_agent(iwe + idx, wv);
        fadd_agent(iwt + idx, wv * ts);
      }
    }
  }
}

// ---------------------------------------------------------------------------
// Wave-32 full reduction using V_WMMA_F32_16X16X4_F32 with B = ones(4x16).
// A[m,0]=s_m (lanes 0-15), A[m,2]=s_{m+16} (lanes 16-31), A[*,1]=A[*,3]=0.
// D[m,n] = s_m + s_{m+16}; per-lane sum of 8 D VGPRs gives half-sums;
// shfl_xor(16) completes the wave total. Exact f32, EXEC must be all ones.
// ---------------------------------------------------------------------------
static __device__ __forceinline__ float wave_reduce_wmma(float s) {
  v2f a;  a.x = s;    a.y = 0.0f;
  v2f bo; bo.x = 1.0f; bo.y = 1.0f;
  v8f c = {};
  c = __builtin_amdgcn_wmma_f32_16x16x4_f32(false, a, false, bo,
                                            (short)0, c, false, false);
  float r = c[0] + c[1] + c[2] + c[3] + c[4] + c[5] + c[6] + c[7];
  r += __shfl_xor(r, 16, 32);
  return r;
}

// ---------------------------------------------------------------------------
// Kernel 2: per-(b,r) tiled reduction of loss numerator and inside count
// grid = B*R*TILES blocks, 256 threads; 4096 pixels per block (16/thread),
// perfectly uniform trip count -> full EXEC at the WMMA.
// ---------------------------------------------------------------------------
__global__ __launch_bounds__(256) void cm_reduce_kernel(
    const float* __restrict__ iwe, const float* __restrict__ iwt,
    float* __restrict__ acc_num, float* __restrict__ acc_cnt) {
  int blk  = blockIdx.x;
  int tile = blk % TILES;
  int br   = blk / TILES;           // b * RREF + r
  int b    = br / RREF;
  int r    = br % RREF;

  int p0 = ((b * 2 + 0) * RREF + r) * IMG;
  int p1 = ((b * 2 + 1) * RREF + r) * IMG;
  int pstart = tile * (IMG / TILES);     // 4096 pixels per tile

  float lsum = 0.0f;
  float csum = 0.0f;
  for (int i = threadIdx.x; i < IMG / TILES; i += 256) {
    int px = pstart + i;
    float e0 = iwe[p0 + px];
    float e1 = iwe[p1 + px];
    float t0 = iwt[p0 + px];
    float t1 = iwt[p1 + px];
    float a0 = t0 / (e0 + EPSF);
    float a1 = t1 / (e1 + EPSF);
    lsum += a0 * a0 + a1 * a1;
    csum += ((e0 + e1) > 0.0f) ? 1.0f : 0.0f;
  }

  float wl = wave_reduce_wmma(lsum);
  float wc = wave_reduce_wmma(csum);

  if ((threadIdx.x & 31) == 0) {
    fadd_agent(acc_num + br, wl);
    fadd_agent(acc_cnt + br, wc);
  }
}

// ---------------------------------------------------------------------------
// Kernel 3: finalize loss = num / (cnt + eps)
// ---------------------------------------------------------------------------
__global__ void cm_finalize_kernel(const float* __restrict__ acc_num,
                                   const float* __restrict__ acc_cnt,
                                   float* __restrict__ out) {
  int i = threadIdx.x;
  if (i < BATCH * RREF) out[i] = acc_num[i] / (acc_cnt[i] + EPSF);
}

// ---------------------------------------------------------------------------
extern "C" void kernel_launch(void* const* d_in, const int* in_sizes, int n_in,
                              void* d_out, int out_size, void* d_ws, size_t ws_size,
                              hipStream_t stream) {
  const float* events = (const float*)d_in[0];  // [B, N, 5]
  const float* flow   = (const float*)d_in[1];  // [B, BASE, H, W, 2]
  float* out = (float*)d_out;                   // [B, R] = 22 floats

  float* ws      = (float*)d_ws;
  float* iwe     = ws;                  // SZ floats
  float* iwt     = ws + SZ;             // SZ floats
  float* acc_num = ws + 2 * SZ;         // 22 (+pad)
  float* acc_cnt = acc_num + 32;

  int ztotal = 2 * SZ + 64;
  cm_zero_kernel<<<2048, 256, 0, stream>>>(ws, ztotal);

  cm_scatter_kernel<<<(BATCH * NEV) / 256, 256, 0, stream>>>(events, flow, iwe, iwt);

  cm_reduce_kernel<<<BATCH * RREF * TILES, 256, 0, stream>>>(iwe, iwt, acc_num, acc_cnt);

  cm_finalize_kernel<<<1, 32, 0, stream>>>(acc_num, acc_cnt, out);
}